// SimpleHierarchicalSoftmax_75600014344313
// MI455X (gfx1250) — compile-verified
//
#include <hip/hip_runtime.h>
#include <hip/hip_bf16.h>
#include <math.h>

// ---------------------------------------------------------------------------
// Problem constants (from reference)
// ---------------------------------------------------------------------------
#define DIM          256
#define NUM_CLUSTERS 400
#define MAXM         256      // MAX_CLUSTER_SIZE
#define NTOK         2048     // B*S = 4*512
#define NEGV         (-1000000000.0f)

typedef float v2f __attribute__((ext_vector_type(2)));
typedef float v8f __attribute__((ext_vector_type(8)));

// ---------------------------------------------------------------------------
// Workspace layout (all fp32/int32, sizes in elements)
//   [0..1]                float accum: {loss_sum, mask_sum}
//   [4 .. 4+2048)         float cluster_lp[NTOK]
//   [4+2048 .. +2048)     int   target_cluster[NTOK]
// ---------------------------------------------------------------------------
#define WS_ACCUM_OFF   0
#define WS_CLP_OFF     4
#define WS_TC_OFF      (4 + NTOK)

// ---------------------------------------------------------------------------
// Kernel 0: zero the accumulators (ws is poisoned by harness, never reset)
// ---------------------------------------------------------------------------
__global__ void hs_init_kernel(float* ws) {
    ws[WS_ACCUM_OFF + 0] = 0.0f;
    ws[WS_ACCUM_OFF + 1] = 0.0f;
}

// ---------------------------------------------------------------------------
// Kernel 1: cluster-level logits via fp32 WMMA + per-token log-softmax.
// One wave (32 lanes) per block handles 16 tokens.
//   A tile: 16 tokens x 4 dims  (v2f per lane: lane m=lane&15 holds rows,
//           lane>=16 holds K+2..K+3 -- per ISA 32-bit A layout)
//   B tile: 4 dims x 16 clusters (symmetric layout)
//   C/D:    16x16 f32, 8 VGPRs
// ---------------------------------------------------------------------------
__global__ __launch_bounds__(32) void hs_cluster_kernel(
    const float* __restrict__ hidden,        // [NTOK, DIM]
    const float* __restrict__ cemb,          // [NUM_CLUSTERS, DIM]
    const int*   __restrict__ targets,       // [NTOK]
    const int*   __restrict__ cassign,       // [NUM_ITEMS]
    float*       __restrict__ cluster_lp,    // ws
    int*         __restrict__ tclust)        // ws
{
    __shared__ alignas(16) float sh_hid[16 * DIM];        // 16 KB
    __shared__ float sh_log[16 * NUM_CLUSTERS];           // 25.6 KB

    const int lane = threadIdx.x;       // 0..31, full wave (EXEC all ones)
    const int tok0 = blockIdx.x * 16;

    // Stage 16 hidden rows into LDS (float4, coalesced).
    {
        const float4* src = (const float4*)(hidden + (size_t)tok0 * DIM);
        float4*       dst = (float4*)sh_hid;
        for (int i = lane; i < 16 * (DIM / 4); i += 32) dst[i] = src[i];
    }
    __syncthreads();

    const int mrow  = lane & 15;        // token row within tile / cluster col
    const int kb    = (lane >> 4) * 2;  // K sub-offset for this half-wave
    const float* arow = sh_hid + mrow * DIM;

    for (int ct = 0; ct < NUM_CLUSTERS / 16; ++ct) {
        const float* brow = cemb + (size_t)(ct * 16 + mrow) * DIM;
        v8f acc = {};
        for (int k = 0; k < DIM; k += 4) {
            v2f a, b;
            a.x = arow[k + kb];
            a.y = arow[k + kb + 1];
            b.x = brow[k + kb];
            b.y = brow[k + kb + 1];
            acc = __builtin_amdgcn_wmma_f32_16x16x4_f32(
                false, a, false, b, (short)0, acc, false, false);
        }
        // D layout: lane col n = lane&15; VGPR r -> row r (+8 for lanes>=16)
        const int col   = ct * 16 + (lane & 15);
        const int rbase = (lane >> 4) * 8;
        #pragma unroll
        for (int r = 0; r < 8; ++r)
            sh_log[(rbase + r) * NUM_CLUSTERS + col] = acc[r];
    }
    __syncthreads();

    // Per-token log-softmax over 400 clusters; lanes 0..15 each own a token.
    if (lane < 16) {
        const int token = tok0 + lane;
        const float* row = sh_log + lane * NUM_CLUSTERS;
        float mx = -INFINITY;
        for (int c = 0; c < NUM_CLUSTERS; ++c) mx = fmaxf(mx, row[c]);
        float s = 0.0f;
        for (int c = 0; c < NUM_CLUSTERS; ++c) s += expf(row[c] - mx);
        const int tgt = targets[token];
        const int tc  = cassign[tgt];
        cluster_lp[token] = row[tc] - mx - logf(s);
        tclust[token]     = tc;
    }
}

// ---------------------------------------------------------------------------
// Kernel 2: item-level masked log-softmax within the target cluster.
// One block of 256 threads per token; thread j owns member slot j.
// ---------------------------------------------------------------------------
__global__ __launch_bounds__(256) void hs_item_kernel(
    const float* __restrict__ hidden,          // [NTOK, DIM]
    const float* __restrict__ iemb,            // [NUM_ITEMS, DIM]
    const int*   __restrict__ cluster_indices, // [NUM_CLUSTERS, MAXM]
    const int*   __restrict__ targets,         // [NTOK]
    const float* __restrict__ loss_mask,       // [NTOK]
    const float* __restrict__ cluster_lp,      // ws
    const int*   __restrict__ tclust,          // ws
    float*       __restrict__ accum)           // ws: {loss_sum, mask_sum}
{
    __shared__ alignas(16) float sh_hid[DIM];
    __shared__ float sred[256];
    __shared__ int   s_pos;
    __shared__ float s_tlp;

    const int j     = threadIdx.x;
    const int token = blockIdx.x;

    sh_hid[j] = hidden[(size_t)token * DIM + j];
    if (j == 0) s_pos = 1 << 30;
    __syncthreads();

    const int  tc     = tclust[token];
    const int  member = cluster_indices[(size_t)tc * MAXM + j];
    const bool valid  = (member != -1);

    float x = NEGV;
    if (valid) {
        const float4* e = (const float4*)(iemb + (size_t)member * DIM);
        const float4* h = (const float4*)sh_hid;
        float acc = 0.0f;
        #pragma unroll
        for (int k = 0; k < DIM / 4; ++k) {
            const float4 ev = e[k];
            const float4 hv = h[k];
            acc = fmaf(ev.x, hv.x, acc);
            acc = fmaf(ev.y, hv.y, acc);
            acc = fmaf(ev.z, hv.z, acc);
            acc = fmaf(ev.w, hv.w, acc);
        }
        x = acc;
    }

    // Block max reduction.
    sred[j] = x;
    __syncthreads();
    for (int off = 128; off > 0; off >>= 1) {
        if (j < off) sred[j] = fmaxf(sred[j], sred[j + off]);
        __syncthreads();
    }
    const float mx = sred[0];
    __syncthreads();

    // Block sum-of-exp reduction.
    const float p = expf(x - mx);
    sred[j] = p;
    __syncthreads();
    for (int off = 128; off > 0; off >>= 1) {
        if (j < off) sred[j] += sred[j + off];
        __syncthreads();
    }
    const float ssum = sred[0];

    const float lp = valid ? (x - mx - logf(ssum)) : 0.0f;

    // target_position = argmax(member == target) -> first match, else 0.
    const int tgt = targets[token];
    if (member == tgt) atomicMin(&s_pos, j);
    __syncthreads();
    const int pos = (s_pos == (1 << 30)) ? 0 : s_pos;
    if (j == pos) s_tlp = lp;
    __syncthreads();

    if (j == 0) {
        const float lm = loss_mask[token];
        atomicAdd(&accum[0], -(cluster_lp[token] + s_tlp) * lm);
        atomicAdd(&accum[1], lm);
    }
}

// ---------------------------------------------------------------------------
// Kernel 3: finalize scalar loss.
// ---------------------------------------------------------------------------
__global__ void hs_final_kernel(const float* __restrict__ accum,
                                float* __restrict__ out) {
    out[0] = accum[0] / (accum[1] + 1e-8f);
}

// ---------------------------------------------------------------------------
// Launcher
// ---------------------------------------------------------------------------
extern "C" void kernel_launch(void* const* d_in, const int* in_sizes, int n_in,
                              void* d_out, int out_size, void* d_ws, size_t ws_size,
                              hipStream_t stream) {
    const float* hidden          = (const float*)d_in[0];  // [4,512,256]
    const float* item_emb        = (const float*)d_in[1];  // [100000,256]
    const float* cluster_emb     = (const float*)d_in[2];  // [400,256]
    const int*   targets         = (const int*)  d_in[3];  // [4,512]
    const float* loss_mask       = (const float*)d_in[4];  // [4,512]
    const int*   cluster_assign  = (const int*)  d_in[5];  // [100000]
    const int*   cluster_indices = (const int*)  d_in[6];  // [400,256]

    float* ws         = (float*)d_ws;
    float* accum      = ws + WS_ACCUM_OFF;
    float* cluster_lp = ws + WS_CLP_OFF;
    int*   tclust     = (int*)(ws + WS_TC_OFF);
    float* out        = (float*)d_out;

    hs_init_kernel<<<1, 1, 0, stream>>>(ws);
    hs_cluster_kernel<<<NTOK / 16, 32, 0, stream>>>(
        hidden, cluster_emb, targets, cluster_assign, cluster_lp, tclust);
    hs_item_kernel<<<NTOK, 256, 0, stream>>>(
        hidden, item_emb, cluster_indices, targets, loss_mask,
        cluster_lp, tclust, accum);
    hs_final_kernel<<<1, 1, 0, stream>>>(accum, out);
}